// Cache_37641093382851
// MI455X (gfx1250) — compile-verified
//
#include <hip/hip_runtime.h>
#include <hip/hip_bf16.h>
#include <math.h>

typedef float v2f __attribute__((ext_vector_type(2)));
typedef float v8f __attribute__((ext_vector_type(8)));

#define B_ROWS   32
#define DK       512
#define DV       512
#define L_DIM    20
#define TOPK     8
#define NTILE    32
#define QSTRIDE  516   // 512 + 4 floats pad: lane m -> bank m*4 (mod 64), conflict-free

// ---------------------------------------------------------------------------
// Kernel 1: scores[b][n] = (q[b] . k[n]) * rsqrt(DK) via V_WMMA_F32_16X16X4_F32
// Block = 128 threads = 4 waves; each wave owns a 16(M)x16(N) tile.
// ---------------------------------------------------------------------------
__global__ __launch_bounds__(128) void scores_wmma_kernel(
    const float* __restrict__ query, const float* __restrict__ keys,
    float* __restrict__ scores, int N)
{
  __shared__ float qlds[B_ROWS * QSTRIDE];
  const int tid = threadIdx.x;

  // Stage full 32x512 query block into LDS, coalesced.
  for (int i = tid; i < B_ROWS * DK; i += 128) {
    int r = i >> 9, c = i & (DK - 1);
    qlds[r * QSTRIDE + c] = query[i];
  }
  __syncthreads();

  const int wave  = tid >> 5;
  const int lane  = tid & 31;
  const int mtile = wave & 1;    // query rows 0..15 or 16..31
  const int ntile = wave >> 1;   // key-column subtile within the block tile
  const int hi    = lane >> 4;   // lane half selects K pair (ISA 7.12.2 layout)
  const int l15   = lane & 15;

  const int n0 = blockIdx.x * NTILE + ntile * 16;
  int n = n0 + l15;
  if (n >= N) n = N - 1;         // clamp (N is a multiple of 32 here)

  const float2* qp   = reinterpret_cast<const float2*>(qlds);
  const int     m    = mtile * 16 + l15;
  const float*  krow = keys + (size_t)n * DK;

  v8f acc = {};
#pragma unroll 8
  for (int k = 0; k < DK; k += 4) {
    const int kk = k + 2 * hi;
    // One prefetch per 64B cacheline per lane (~1KB ahead), not per load.
    if ((k & 15) == 0)
      __builtin_prefetch(krow + kk + 256, 0, 1);   // global_prefetch_b8
    // A fragment: Q[m][kk], Q[m][kk+1] from LDS (8B aligned, bank-conflict free)
    float2 af = qp[(m * QSTRIDE + kk) >> 1];
    // B fragment: keys[n][kk], keys[n][kk+1]; lanes l and l+16 share row n,
    // so the wave streams 16 key rows contiguously.
    float2 bf = *reinterpret_cast<const float2*>(krow + kk);
    v2f a; a[0] = af.x; a[1] = af.y;
    v2f b; b[0] = bf.x; b[1] = bf.y;
    acc = __builtin_amdgcn_wmma_f32_16x16x4_f32(false, a, false, b,
                                                (short)0, acc, false, false);
  }

  // C/D layout: VGPR r -> M = r (lanes 0..15) / M = 8+r (lanes 16..31)
  const float scale = 0.04419417382415922f;  // 1/sqrt(512)
  const int col = n0 + l15;
  if (col < N) {
    for (int r = 0; r < 8; ++r) {
      int row = mtile * 16 + hi * 8 + r;
      scores[(size_t)row * N + col] = acc[r] * scale;
    }
  }
}

// ---------------------------------------------------------------------------
// Kernel 2: per batch row: max, sum(exp), top-8 -> weights to d_out, idx to ws
// ---------------------------------------------------------------------------
__global__ __launch_bounds__(256) void softmax_topk_kernel(
    const float* __restrict__ scores, float* __restrict__ out_w,
    int* __restrict__ out_idx, int N)
{
  __shared__ float sred[256];
  __shared__ int   sidx[256];
  __shared__ float cs[256 * TOPK];
  __shared__ int   ci[256 * TOPK];

  const int b   = blockIdx.x;
  const int tid = threadIdx.x;
  const float* row = scores + (size_t)b * N;
  const float NEG = -3.402823466e38f;

  // pass 1: row max
  float m = NEG;
  for (int n = tid; n < N; n += 256) m = fmaxf(m, row[n]);
  sred[tid] = m; __syncthreads();
  for (int s = 128; s > 0; s >>= 1) {
    if (tid < s) sred[tid] = fmaxf(sred[tid], sred[tid + s]);
    __syncthreads();
  }
  const float rowmax = sred[0];
  __syncthreads();

  // pass 2: sum of exp + per-thread top-8 (sorted insertion)
  float ts[TOPK]; int ti[TOPK];
  for (int j = 0; j < TOPK; ++j) { ts[j] = NEG; ti[j] = 0; }
  float sum = 0.f;
  for (int n = tid; n < N; n += 256) {
    float v = row[n];
    sum += __expf(v - rowmax);
    if (v > ts[TOPK - 1]) {
      int j = TOPK - 1;
      while (j > 0 && ts[j - 1] < v) { ts[j] = ts[j - 1]; ti[j] = ti[j - 1]; --j; }
      ts[j] = v; ti[j] = n;
    }
  }
  sred[tid] = sum; __syncthreads();
  for (int s = 128; s > 0; s >>= 1) {
    if (tid < s) sred[tid] += sred[tid + s];
    __syncthreads();
  }
  const float sumexp = sred[0];
  __syncthreads();

  // candidate pool: 256 threads x 8 candidates
  for (int j = 0; j < TOPK; ++j) {
    cs[tid * TOPK + j] = ts[j];
    ci[tid * TOPK + j] = ti[j];
  }
  __syncthreads();

  // 8 argmax rounds over the 2048-entry pool
  for (int r = 0; r < TOPK; ++r) {
    float lv = NEG; int lp = 0;
    for (int j = tid * TOPK; j < tid * TOPK + TOPK; ++j)
      if (cs[j] > lv) { lv = cs[j]; lp = j; }
    sred[tid] = lv; sidx[tid] = lp; __syncthreads();
    for (int s = 128; s > 0; s >>= 1) {
      if (tid < s && sred[tid + s] > sred[tid]) {
        sred[tid] = sred[tid + s]; sidx[tid] = sidx[tid + s];
      }
      __syncthreads();
    }
    if (tid == 0) {
      int p = sidx[0];
      out_w[b * TOPK + r]  = __expf(cs[p] - rowmax) / sumexp;
      out_idx[b * TOPK + r] = ci[p];
      cs[p] = NEG;   // remove winner
    }
    __syncthreads();
  }
}

// ---------------------------------------------------------------------------
// Kernel 3: gather values[idx] rows (20x512 f32 = 40KB each) with B128 copies
// ---------------------------------------------------------------------------
__global__ __launch_bounds__(256) void gather_kernel(
    const float* __restrict__ values, const int* __restrict__ idx,
    float* __restrict__ out)
{
  const int rowsel = idx[blockIdx.x];
  const float4* src = reinterpret_cast<const float4*>(
      values + (size_t)rowsel * (L_DIM * DV));
  float4* dst = reinterpret_cast<float4*>(
      out + (size_t)blockIdx.x * (L_DIM * DV));
  const int n4 = (L_DIM * DV) / 4;   // 2560 float4
  for (int i = threadIdx.x; i < n4; i += 256) dst[i] = src[i];
}

// ---------------------------------------------------------------------------
extern "C" void kernel_launch(void* const* d_in, const int* in_sizes, int n_in,
                              void* d_out, int out_size, void* d_ws, size_t ws_size,
                              hipStream_t stream) {
  const float* query  = (const float*)d_in[0];   // [32, 512]
  const float* keys   = (const float*)d_in[1];   // [N, 512]
  const float* values = (const float*)d_in[2];   // [N, 20, 512]
  const int N = in_sizes[1] / DK;                // 20000

  float* out      = (float*)d_out;               // [256] weights ++ [256,20,512]
  float* scores   = (float*)d_ws;                // 32*N floats
  int*   topk_idx = (int*)((char*)d_ws + (size_t)B_ROWS * N * sizeof(float));

  const int nblocks = (N + NTILE - 1) / NTILE;   // 625
  scores_wmma_kernel<<<nblocks, 128, 0, stream>>>(query, keys, scores, N);
  softmax_topk_kernel<<<B_ROWS, 256, 0, stream>>>(scores, out, topk_idx, N);
  gather_kernel<<<B_ROWS * TOPK, 256, 0, stream>>>(values, topk_idx,
                                                   out + B_ROWS * TOPK);
}